// QuantityInjector_30331059044436
// MI455X (gfx1250) — compile-verified
//
#include <hip/hip_runtime.h>
#include <hip/hip_bf16.h>

// Problem constants (from the reference)
#define VOCAB 30522
#define B_    64
#define S_    1024
#define J_    768
#define JE_   77
#define JM_   691
#define N_    4096            // B * S/STRIDE
#define NUM_ID 1
#define JTILES 44             // ceil(691/16)
#define NTILES 256            // 4096/16

typedef __attribute__((ext_vector_type(2))) float v2f;
typedef __attribute__((ext_vector_type(4))) float v4f;
typedef __attribute__((ext_vector_type(8))) float v8f;

// ---------------------------------------------------------------------------
// Kernel 1: embedding gather. One 192-thread block (6 waves) per output row.
// ids[blockIdx.x] is wave-uniform -> scalar load; each lane does one b128
// load from the table (L2-resident, 89.5 MiB < 192 MB L2) and one
// NON-TEMPORAL b128 store of the 192 MiB output so the table stays cached.
// ---------------------------------------------------------------------------
__global__ __launch_bounds__(192) void qi_gather_kernel(
    const float* __restrict__ table, const int* __restrict__ ids,
    float* __restrict__ out) {
  const int row = blockIdx.x;              // 0 .. B*S-1
  const int id  = ids[row];                // uniform -> s_load
  const v4f* src = (const v4f*)table + (size_t)id  * (J_ / 4);
  v4f*       dst = (v4f*)out        + (size_t)row * (J_ / 4);
  v4f v = src[threadIdx.x];
  __builtin_nontemporal_store(v, &dst[threadIdx.x]);
}

// ---------------------------------------------------------------------------
// Kernel 2: exponent-embedding part of the number rows (columns 0..76).
// out[pos_b[n], pos_t[n], j] = embed[NUM_ID, j] + exp_embed[clip(e)+20, j]
// 4096*77 elements total -> negligible.
// ---------------------------------------------------------------------------
__global__ __launch_bounds__(256) void qi_exp_kernel(
    const float* __restrict__ embed, const float* __restrict__ exp_embed,
    const int* __restrict__ pos_b, const int* __restrict__ pos_t,
    const int* __restrict__ span_e, float* __restrict__ out) {
  int i = blockIdx.x * blockDim.x + threadIdx.x;
  if (i >= N_ * JE_) return;
  int n = i / JE_;
  int j = i - n * JE_;
  int e = span_e[n];
  e = (e < -20) ? -20 : (e > 20 ? 20 : e);
  float v = embed[NUM_ID * J_ + j] + exp_embed[(e + 20) * JE_ + j];
  size_t off = (size_t)pos_b[n] * (S_ * J_) + (size_t)pos_t[n] * J_ + j;
  out[off] = v;
}

// ---------------------------------------------------------------------------
// Kernel 3: Gaussian mantissa part (columns 77..767) via V_WMMA_F32_16X16X4_F32.
// One wave computes a 16(spans) x 16(protos) tile of -(m_i - p_j)^2 as the
// K=4 matmul  [m^2, m, 1, 0] x [-1, 2p, -p^2, 0]^T , then exp() + base_num
// and scatter-stores into the planted rows.
//
// VGPR layouts (CDNA5 ISA 7.12.2, wave32):
//   A 16x4 f32 : lanes 0-15 -> M=lane, {v0=K0, v1=K1}; lanes 16-31 -> {K2, K3}
//   B 4x16 f32 : lanes 0-15 -> N=lane, {v0=K0 row, v1=K1 row}; lanes 16-31 -> {K2, K3}
//   D 16x16 f32: vgpr r, lanes 0-15 -> (M=r,   N=lane)
//                        lanes 16-31-> (M=r+8, N=lane-16)
// ---------------------------------------------------------------------------
__global__ __launch_bounds__(256) void qi_mant_wmma_kernel(
    const float* __restrict__ embed,
    const int* __restrict__ pos_b, const int* __restrict__ pos_t,
    const float* __restrict__ mant, float* __restrict__ out) {
  const int wave = blockIdx.x * (blockDim.x >> 5) + (threadIdx.x >> 5);
  const int lane = threadIdx.x & 31;
  const int half = lane >> 4;        // 0: lanes 0-15, 1: lanes 16-31
  const int lm   = lane & 15;

  const int ntile = wave / JTILES;
  const int jtile = wave - ntile * JTILES;
  const int nbase = ntile << 4;      // first span of this tile
  const int jm    = (jtile << 4) + lm;  // proto index handled by this lane (N)

  // A operand: row M = lm is [m^2, m, 1, 0]
  const float m = mant[nbase + lm];
  // B operand: col N = lm is [-1, 2p, -p^2, 0]
  const float p = -10.0f + (20.0f / 690.0f) * (float)jm;

  v2f a, b;
  if (half == 0) { a.x = m * m;  a.y = m;          b.x = -1.0f;  b.y = 2.0f * p; }
  else           { a.x = 1.0f;   a.y = 0.0f;       b.x = -p * p; b.y = 0.0f;     }

  v8f c = {};
  // D[i][j] = -m_i^2 + 2*m_i*p_j - p_j^2 = -(m_i - p_j)^2
  c = __builtin_amdgcn_wmma_f32_16x16x4_f32(
      /*neg_a=*/false, a, /*neg_b=*/false, b,
      /*c_mod=*/(short)0, c, /*reuse_a=*/false, /*reuse_b=*/false);

  if (jm < JM_) {
    const int col = JE_ + jm;                    // output column 77..767
    const float bn = embed[NUM_ID * J_ + col];   // base_num[col]
#pragma unroll
    for (int r = 0; r < 8; ++r) {
      const int n = nbase + (half << 3) + r;     // span index (row M)
      size_t off = (size_t)pos_b[n] * (S_ * J_) + (size_t)pos_t[n] * J_ + col;
      out[off] = bn + __expf(c[r]);
    }
  }
}

// ---------------------------------------------------------------------------
extern "C" void kernel_launch(void* const* d_in, const int* in_sizes, int n_in,
                              void* d_out, int out_size, void* d_ws, size_t ws_size,
                              hipStream_t stream) {
  const float* embed_table = (const float*)d_in[0];  // [VOCAB, 768]
  const float* exp_embed   = (const float*)d_in[1];  // [41, 77]
  const int*   input_ids   = (const int*)  d_in[2];  // [64, 1024]
  const int*   pos_b       = (const int*)  d_in[3];  // [4096]
  const int*   pos_t       = (const int*)  d_in[4];  // [4096]
  const int*   span_e      = (const int*)  d_in[5];  // [4096]
  const float* span_m      = (const float*)d_in[6];  // [4096]
  float*       out         = (float*)d_out;          // [64, 1024, 768]

  // 1) full gather: one block per token row
  qi_gather_kernel<<<B_ * S_, 192, 0, stream>>>(embed_table, input_ids, out);

  // 2) exponent-embedding columns of the planted rows
  const int exp_elems = N_ * JE_;
  qi_exp_kernel<<<(exp_elems + 255) / 256, 256, 0, stream>>>(
      embed_table, exp_embed, pos_b, pos_t, span_e, out);

  // 3) Gaussian mantissa columns via WMMA: 256 n-tiles x 44 j-tiles = 11264
  //    waves, 8 waves per 256-thread block -> 1408 blocks (exact, no tail).
  qi_mant_wmma_kernel<<<(NTILES * JTILES) / 8, 256, 0, stream>>>(
      embed_table, pos_b, pos_t, span_m, out);
}